// GNN_33672543600993
// MI455X (gfx1250) — compile-verified
//
#include <hip/hip_runtime.h>
#include <cstdint>
#include <cstddef>

// ---------------------------------------------------------------------------
// GATConv-style GNN forward for MI455X (gfx1250), wave32.
//  * GEMMs (h@Wl, h@Wr) use V_WMMA_F32_16X16X4_F32 (exact fp32, matrix pipe).
//  * Edge logits fuse ea@We (We staged in LDS) + leaky_relu + att dot.
//  * Scatter-softmax/aggregation via CSR-by-dst (no atomics in hot loops).
// ---------------------------------------------------------------------------

#define NN   20000        // nodes
#define NE   80000        // edges (no self loops)
#define NEF  (NN + NE)    // edges + self loops
#define NG   400          // graphs
#define HIDD 512

typedef float v2f __attribute__((ext_vector_type(2)));
typedef float v8f __attribute__((ext_vector_type(8)));

// ------------------------------ utilities ----------------------------------
__global__ void k_zero_u32(uint32_t* __restrict__ p, long n) {
  long i = (long)blockIdx.x * blockDim.x + threadIdx.x;
  if (i < n) p[i] = 0u;
}

// ------------------------- feature construction ----------------------------
// h0[n, 0:64]=atom_emb[id] (row0 zeroed), [64:72]=x[:,1:9], [72:74]=bool_emb,
// [74:80]=0 padding so layer-0 K is a multiple of 16 for WMMA.
__global__ void k_build_h(const float* __restrict__ x,
                          const float* __restrict__ atom_emb,
                          const float* __restrict__ bool_emb,
                          float* __restrict__ h) {
  long i = (long)blockIdx.x * blockDim.x + threadIdx.x;
  if (i >= (long)NN * 80) return;
  int n = (int)(i / 80), c = (int)(i % 80);
  float v;
  if (c < 64) {
    int id = (int)x[n * 10 + 0];
    v = (id == 0) ? 0.f : atom_emb[id * 64 + c];
  } else if (c < 72) {
    v = x[n * 10 + (c - 63)];              // x cols 1..8
  } else if (c < 74) {
    int b = (int)x[n * 10 + 9];
    v = (b == 2) ? 0.f : bool_emb[b * 2 + (c - 72)];
  } else {
    v = 0.f;
  }
  h[(size_t)n * 80 + c] = v;
}

// ea_full[e,0:16]=bond_emb[id], [16]=f, [17:19]=bool(conj), [19:21]=bool(arom).
// Also accumulates per-dst sums/counts for self-loop mean attrs and copies
// src/dst into the extended arrays.
__global__ void k_build_edges(const float* __restrict__ edge_attr,
                              const float* __restrict__ bond_emb,
                              const float* __restrict__ bool_emb,
                              const int* __restrict__ edge_index,
                              float* __restrict__ eaf,
                              float* __restrict__ loop_sum,
                              int* __restrict__ cnt_in,
                              int* __restrict__ srcF, int* __restrict__ dstF) {
  int e = blockIdx.x * blockDim.x + threadIdx.x;
  if (e >= NE) return;
  int s = edge_index[e], d = edge_index[NE + e];
  srcF[e] = s; dstF[e] = d;
  float w[21];
  int bid = (int)edge_attr[e * 4 + 0];
#pragma unroll
  for (int c = 0; c < 16; ++c) w[c] = (bid == 0) ? 0.f : bond_emb[bid * 16 + c];
  w[16] = edge_attr[e * 4 + 1];
  int cj = (int)edge_attr[e * 4 + 2];
  w[17] = (cj == 2) ? 0.f : bool_emb[cj * 2 + 0];
  w[18] = (cj == 2) ? 0.f : bool_emb[cj * 2 + 1];
  int ar = (int)edge_attr[e * 4 + 3];
  w[19] = (ar == 2) ? 0.f : bool_emb[ar * 2 + 0];
  w[20] = (ar == 2) ? 0.f : bool_emb[ar * 2 + 1];
#pragma unroll
  for (int c = 0; c < 21; ++c) {
    eaf[(size_t)e * 21 + c] = w[c];
    atomicAdd(&loop_sum[(size_t)d * 21 + c], w[c]);
  }
  atomicAdd(&cnt_in[d], 1);
}

// Self-loop attrs (mean of incoming), self-loop src/dst, CSR degree (+1 loop).
__global__ void k_finalize_nodes(const float* __restrict__ loop_sum,
                                 const int* __restrict__ cnt_in,
                                 float* __restrict__ eaf,
                                 int* __restrict__ srcF, int* __restrict__ dstF,
                                 int* __restrict__ deg) {
  int n = blockIdx.x * blockDim.x + threadIdx.x;
  if (n >= NN) return;
  int c0 = cnt_in[n];
  float inv = 1.0f / (float)(c0 > 1 ? c0 : 1);
#pragma unroll
  for (int c = 0; c < 21; ++c)
    eaf[(size_t)(NE + n) * 21 + c] = loop_sum[(size_t)n * 21 + c] * inv;
  srcF[NE + n] = n; dstF[NE + n] = n;
  deg[n] = c0 + 1;
}

// Single-block exclusive scan (20000 ints) -> CSR row offsets.
__global__ void k_scan_excl(const int* __restrict__ deg, int* __restrict__ offs,
                            int n) {
  __shared__ int sh[1024];
  __shared__ int carry;
  if (threadIdx.x == 0) { carry = 0; offs[0] = 0; }
  __syncthreads();
  for (int base = 0; base < n; base += 1024) {
    int i = base + (int)threadIdx.x;
    sh[threadIdx.x] = (i < n) ? deg[i] : 0;
    __syncthreads();
    for (int ofs = 1; ofs < 1024; ofs <<= 1) {
      int t = (threadIdx.x >= (unsigned)ofs) ? sh[threadIdx.x - ofs] : 0;
      __syncthreads();
      sh[threadIdx.x] += t;
      __syncthreads();
    }
    int incl = sh[threadIdx.x] + carry;
    if (i < n) offs[i + 1] = incl;
    __syncthreads();
    if (threadIdx.x == 0) carry += sh[1023];
    __syncthreads();
  }
}

__global__ void k_fill_csr(const int* __restrict__ dstF,
                           const int* __restrict__ offs,
                           int* __restrict__ cursor, int* __restrict__ csr) {
  int e = blockIdx.x * blockDim.x + threadIdx.x;
  if (e >= NEF) return;
  int d = dstF[e];
  int p = atomicAdd(&cursor[d], 1);
  csr[offs[d] + p] = e;
}

// ------------------------------ WMMA GEMM ----------------------------------
// C[M x Nc] = A[M x Kpad] (lda=Kpad, zero-padded rows of K beyond Kreal on the
// W side) * W[Kreal x Nc] + bias.  Block = 128 threads (4 waves), 64x64 tile,
// K staged in 16-deep LDS chunks.  fp32 WMMA: A frag 16x4, B frag 4x16.
__global__ void k_gemm_wmma_f32(const float* __restrict__ A,
                                const float* __restrict__ W,
                                const float* __restrict__ bias,
                                float* __restrict__ C,
                                int M, int Kpad, int Kreal, int Nc) {
  __shared__ float aS[16 * 80];   // k-major A tile: aS[k*80 + m], m<64
  __shared__ float bS[16 * 80];   // row-major B tile: bS[k*80 + n], n<64
  const int tid = threadIdx.x;
  const int wv = tid >> 5;            // wave 0..3 -> 16-row strip
  const int lane = tid & 31;
  const int l16 = lane & 15;
  const int khalf = (lane < 16) ? 0 : 2;   // fp32 A/B frag K split per lane half
  const int row0 = blockIdx.y * 64;
  const int col0 = blockIdx.x * 64;

  v8f acc[4] = {};

  for (int kc = 0; kc < Kpad; kc += 16) {
    // A tile (64 rows x 16 k), stored k-major; stride 80 keeps the half-wave
    // fragment reads on disjoint bank halves (2*80 % 64 == 32).
#pragma unroll
    for (int it = 0; it < 8; ++it) {
      int i = it * 128 + tid;               // 0..1023
      int m = i >> 4, k = i & 15;
      int gr = row0 + m;
      aS[k * 80 + m] = (gr < M) ? A[(size_t)gr * Kpad + kc + k] : 0.f;
    }
    // B tile (16 k x 64 cols)
#pragma unroll
    for (int it = 0; it < 8; ++it) {
      int i = it * 128 + tid;
      int n = i & 63, k = i >> 6;
      int gk = kc + k;
      bS[k * 80 + n] = (gk < Kreal) ? W[(size_t)gk * Nc + col0 + n] : 0.f;
    }
    __syncthreads();
#pragma unroll
    for (int kk = 0; kk < 16; kk += 4) {
      const int ka = kk + khalf;
      v2f a;                                 // A 16x4: lanes0-15 K=0,1; 16-31 K=2,3
      a.x = aS[ka * 80 + (wv << 4) + l16];
      a.y = aS[(ka + 1) * 80 + (wv << 4) + l16];
#pragma unroll
      for (int t = 0; t < 4; ++t) {
        v2f b;                               // B 4x16, same half-lane K split
        b.x = bS[ka * 80 + (t << 4) + l16];
        b.y = bS[(ka + 1) * 80 + (t << 4) + l16];
        acc[t] = __builtin_amdgcn_wmma_f32_16x16x4_f32(
            false, a, false, b, (short)0, acc[t], false, false);
      }
    }
    __syncthreads();
  }

  // C layout: VGPR r holds M=r (lanes 0-15) / M=8+r (lanes 16-31), N=l16.
  const int rbase = row0 + (wv << 4) + ((lane < 16) ? 0 : 8);
#pragma unroll
  for (int t = 0; t < 4; ++t) {
    int c = col0 + (t << 4) + l16;
    float bv = bias[c];
#pragma unroll
    for (int r = 0; r < 8; ++r) {
      int gr = rbase + r;
      if (gr < M) C[(size_t)gr * Nc + c] = acc[t][r] + bv;
    }
  }
}

// ----------------------------- edge logits ---------------------------------
// logit[e] = sum_k leaky_relu(xl[src,k] + xr[dst,k] + (ea[e]@We)[k]) * att[k]
// One wave per edge; We (21 x dout) + att staged in dynamic LDS (<=180 KB,
// fits the 320 KB WGP LDS).  m is never materialized.
__global__ void k_edge_logit(const float* __restrict__ xl,
                             const float* __restrict__ xr,
                             const float* __restrict__ eaf,
                             const float* __restrict__ We,
                             const float* __restrict__ att,
                             const int* __restrict__ srcF,
                             const int* __restrict__ dstF,
                             float* __restrict__ logit, int dout) {
  extern __shared__ float sm[];
  float* WeS = sm;                 // 21*dout
  float* attS = sm + 21 * dout;    // dout
  for (int i = threadIdx.x; i < 21 * dout; i += blockDim.x) WeS[i] = We[i];
  for (int i = threadIdx.x; i < dout; i += blockDim.x) attS[i] = att[i];
  __syncthreads();

  const int lane = threadIdx.x & 31;
  const int wid = blockIdx.x * (blockDim.x >> 5) + (threadIdx.x >> 5);
  const int nw = gridDim.x * (blockDim.x >> 5);
  for (int e = wid; e < NEF; e += nw) {
    int s = srcF[e], d = dstF[e];
    float ea[21];
#pragma unroll
    for (int j = 0; j < 21; ++j) ea[j] = eaf[(size_t)e * 21 + j];  // broadcast
    const float* xls = xl + (size_t)s * dout;
    const float* xrd = xr + (size_t)d * dout;
    float acc = 0.f;
    for (int k = lane; k < dout; k += 32) {
      float ew = 0.f;
#pragma unroll
      for (int j = 0; j < 21; ++j) ew += ea[j] * WeS[j * dout + k];
      float v = xls[k] + xrd[k] + ew;
      v = (v > 0.f) ? v : 0.2f * v;          // leaky_relu(0.2)
      acc += v * attS[k];
    }
#pragma unroll
    for (int m = 16; m >= 1; m >>= 1) acc += __shfl_xor(acc, m, 32);
    if (lane == 0) logit[e] = acc;
  }
}

// ------------------------- per-dst softmax (CSR) ---------------------------
__global__ void k_node_softmax(const float* __restrict__ logit,
                               const int* __restrict__ offs,
                               const int* __restrict__ csr,
                               float* __restrict__ alpha) {
  const int lane = threadIdx.x & 31;
  const int n = blockIdx.x * (blockDim.x >> 5) + (threadIdx.x >> 5);
  if (n >= NN) return;
  const int o0 = offs[n], o1 = offs[n + 1];   // >=1 (self loop)
  float mx = -3.402823466e38f;
  for (int i = o0 + lane; i < o1; i += 32) mx = fmaxf(mx, logit[csr[i]]);
#pragma unroll
  for (int m = 16; m >= 1; m >>= 1) mx = fmaxf(mx, __shfl_xor(mx, m, 32));
  float den = 0.f;
  for (int i = o0 + lane; i < o1; i += 32) den += expf(logit[csr[i]] - mx);
#pragma unroll
  for (int m = 16; m >= 1; m >>= 1) den += __shfl_xor(den, m, 32);
  float inv = 1.0f / den;
  for (int i = o0 + lane; i < o1; i += 32) {
    int e = csr[i];
    alpha[e] = expf(logit[e] - mx) * inv;
  }
}

// --------------------------- aggregation (CSR) -----------------------------
// h[n,k] = relu(b[k] + sum_{e in in(n)} alpha[e] * xl[src[e], k])
__global__ void k_aggregate(const float* __restrict__ xl,
                            const float* __restrict__ alpha,
                            const int* __restrict__ offs,
                            const int* __restrict__ csr,
                            const int* __restrict__ srcF,
                            const float* __restrict__ bias,
                            float* __restrict__ hout, int dout) {
  const int n = blockIdx.x;
  const int k = blockIdx.y * blockDim.x + threadIdx.x;
  if (k >= dout) return;
  const int o0 = offs[n], o1 = offs[n + 1];
  float acc = bias[k];
  for (int i = o0; i < o1; ++i) {
    int e = csr[i];
    acc += alpha[e] * xl[(size_t)srcF[e] * dout + k];   // coalesced rows
  }
  hout[(size_t)n * dout + k] = fmaxf(acc, 0.f);
}

// ------------------------------- pooling -----------------------------------
__global__ void k_pool_count(const int* __restrict__ batch, int* __restrict__ gcnt) {
  int n = blockIdx.x * blockDim.x + threadIdx.x;
  if (n < NN) atomicAdd(&gcnt[batch[n]], 1);
}
__global__ void k_pool_sum(const float* __restrict__ h,
                           const int* __restrict__ batch,
                           float* __restrict__ out) {
  int n = blockIdx.x;
  int k = blockIdx.y * blockDim.x + threadIdx.x;
  if (k >= HIDD) return;
  atomicAdd(&out[(size_t)batch[n] * HIDD + k], h[(size_t)n * HIDD + k]);
}
__global__ void k_pool_div(float* __restrict__ out, const int* __restrict__ gcnt) {
  int i = blockIdx.x * blockDim.x + threadIdx.x;
  if (i >= NG * HIDD) return;
  int g = i / HIDD;
  int c = gcnt[g];
  out[i] /= (float)(c > 1 ? c : 1);
}

// ------------------------------- launcher ----------------------------------
extern "C" void kernel_launch(void* const* d_in, const int* in_sizes, int n_in,
                              void* d_out, int out_size, void* d_ws, size_t ws_size,
                              hipStream_t stream) {
  const float* x        = (const float*)d_in[0];
  const float* edge_att = (const float*)d_in[1];
  const float* atom_emb = (const float*)d_in[2];
  const float* bond_emb = (const float*)d_in[3];
  const float* bool_emb = (const float*)d_in[4];
  const int*   edge_idx = (const int*)d_in[n_in - 2];
  const int*   batch    = (const int*)d_in[n_in - 1];

  static const int LDIN[4]  = {80, 2048, 1024, 512};   // padded lda (A side)
  static const int KREAL[4] = {74, 2048, 1024, 512};   // real K (W side)
  static const int DREAL[4] = {74, 2048, 1024, 512};
  static const int DOUT[4]  = {2048, 1024, 512, 512};

  // Resolve param pointers: either 28 pytree leaves (n_in==35) or one blob.
  const float* Wl[4]; const float* bl[4]; const float* Wr[4]; const float* br[4];
  const float* We[4]; const float* att[4]; const float* bb[4];
  if (n_in >= 35) {
    for (int l = 0; l < 4; ++l) {
      int wi = 5 + 7 * l;
      Wl[l]  = (const float*)d_in[wi + 0];
      bl[l]  = (const float*)d_in[wi + 1];
      Wr[l]  = (const float*)d_in[wi + 2];
      br[l]  = (const float*)d_in[wi + 3];
      We[l]  = (const float*)d_in[wi + 4];
      att[l] = (const float*)d_in[wi + 5];
      bb[l]  = (const float*)d_in[wi + 6];
    }
  } else {
    const float* P = (const float*)d_in[5];
    size_t off = 0;
    for (int l = 0; l < 4; ++l) {
      size_t din = (size_t)DREAL[l], dout = (size_t)DOUT[l];
      Wl[l]  = P + off; off += din * dout;
      bl[l]  = P + off; off += dout;
      Wr[l]  = P + off; off += din * dout;
      br[l]  = P + off; off += dout;
      We[l]  = P + off; off += 21 * dout;
      att[l] = P + off; off += dout;
      bb[l]  = P + off; off += dout;
    }
  }

  // --------------- workspace layout (~505 MB) ---------------
  size_t off = 0;
  auto walloc = [&](size_t bytes) -> void* {
    void* p = (char*)d_ws + off;
    off += (bytes + 255) & ~(size_t)255;
    return p;
  };
  float* hbuf     = (float*)walloc((size_t)NN * 2048 * 4);
  float* xl       = (float*)walloc((size_t)NN * 2048 * 4);
  float* xr       = (float*)walloc((size_t)NN * 2048 * 4);
  float* eaf      = (float*)walloc((size_t)NEF * 21 * 4);
  float* logit    = (float*)walloc((size_t)NEF * 4);
  float* alpha    = (float*)walloc((size_t)NEF * 4);
  int*   srcF     = (int*)walloc((size_t)NEF * 4);
  int*   dstF     = (int*)walloc((size_t)NEF * 4);
  float* loop_sum = (float*)walloc((size_t)NN * 21 * 4);
  int*   cnt_in   = (int*)walloc((size_t)NN * 4);
  int*   deg      = (int*)walloc((size_t)NN * 4);
  int*   offs     = (int*)walloc((size_t)(NN + 1) * 4);
  int*   cursor   = (int*)walloc((size_t)NN * 4);
  int*   csr      = (int*)walloc((size_t)NEF * 4);
  int*   gcnt     = (int*)walloc((size_t)NG * 4);
  (void)ws_size;

  auto zero = [&](void* p, long n32) {
    k_zero_u32<<<(unsigned)((n32 + 255) / 256), 256, 0, stream>>>((uint32_t*)p, n32);
  };
  zero(loop_sum, (long)NN * 21);
  zero(cnt_in, NN);
  zero(cursor, NN);
  zero(gcnt, NG);
  zero(d_out, (long)NG * HIDD);

  // ---- feature build + CSR ----
  k_build_h<<<(unsigned)(((long)NN * 80 + 255) / 256), 256, 0, stream>>>(
      x, atom_emb, bool_emb, hbuf);
  k_build_edges<<<(NE + 255) / 256, 256, 0, stream>>>(
      edge_att, bond_emb, bool_emb, edge_idx, eaf, loop_sum, cnt_in, srcF, dstF);
  k_finalize_nodes<<<(NN + 255) / 256, 256, 0, stream>>>(
      loop_sum, cnt_in, eaf, srcF, dstF, deg);
  k_scan_excl<<<1, 1024, 0, stream>>>(deg, offs, NN);
  k_fill_csr<<<(NEF + 255) / 256, 256, 0, stream>>>(dstF, offs, cursor, csr);

  // ---- 4 GNN layers ----
  for (int l = 0; l < 4; ++l) {
    const int kpad = LDIN[l], kreal = KREAL[l], dout = DOUT[l];
    dim3 ggrid((unsigned)(dout / 64), (unsigned)((NN + 63) / 64));
    k_gemm_wmma_f32<<<ggrid, 128, 0, stream>>>(hbuf, Wl[l], bl[l], xl,
                                               NN, kpad, kreal, dout);
    k_gemm_wmma_f32<<<ggrid, 128, 0, stream>>>(hbuf, Wr[l], br[l], xr,
                                               NN, kpad, kreal, dout);
    size_t shmem = (size_t)(22 * dout) * sizeof(float);  // We + att in LDS
    k_edge_logit<<<1024, 256, shmem, stream>>>(xl, xr, eaf, We[l], att[l],
                                               srcF, dstF, logit, dout);
    k_node_softmax<<<(NN + 7) / 8, 256, 0, stream>>>(logit, offs, csr, alpha);
    dim3 agrid((unsigned)NN, (unsigned)((dout + 255) / 256));
    k_aggregate<<<agrid, 256, 0, stream>>>(xl, alpha, offs, csr, srcF, bb[l],
                                           hbuf, dout);   // hbuf overwritten in place
  }

  // ---- graph mean pooling ----
  k_pool_count<<<(NN + 255) / 256, 256, 0, stream>>>(batch, gcnt);
  k_pool_sum<<<dim3((unsigned)NN, (unsigned)(HIDD / 256)), 256, 0, stream>>>(
      hbuf, batch, (float*)d_out);
  k_pool_div<<<((long)NG * HIDD + 255) / 256, 256, 0, stream>>>((float*)d_out, gcnt);
  (void)in_sizes; (void)out_size;
}